// Involution1D_90013924590112
// MI455X (gfx1250) — compile-verified
//
#include <hip/hip_runtime.h>
#include <hip/hip_bf16.h>

// ---------------------------------------------------------------------------
// Fused Involution1D for MI455X (gfx1250, wave32).
//   B=8, C=512, L=4096, Cr=128, K=7, G=8, Cg=64, PAD=3, KG=56
// One workgroup = one (batch, 64-column) slice. 256 threads = 8 waves.
// GEMMs use V_WMMA_F32_16X16X4_F32 (full fp32 matrix path).
// ---------------------------------------------------------------------------

typedef __attribute__((ext_vector_type(2))) float v2f;
typedef __attribute__((ext_vector_type(8))) float v8f;

// Branch purely on builtin availability at the use site. The pass that has
// the builtin (device) gets real WMMA; any pass without it (host) gets a
// type-correct placeholder that is parsed but never executed.
__device__ __forceinline__ v8f wmma4(v2f a, v2f b, v8f c) {
#if __has_builtin(__builtin_amdgcn_wmma_f32_16x16x4_f32)
    return __builtin_amdgcn_wmma_f32_16x16x4_f32(
        /*neg_a=*/false, a, /*neg_b=*/false, b,
        /*c_mod=*/(short)0, c, /*reuse_a=*/false, /*reuse_b=*/false);
#else
    c[0] += a.x * b.x;   // host-pass parse placeholder only
    return c;
#endif
}

namespace {
constexpr int Bn   = 8;
constexpr int Cn   = 512;
constexpr int Ln   = 4096;
constexpr int Cr   = 128;
constexpr int Kk   = 7;
constexpr int Gn   = 8;
constexpr int Cg   = 64;   // C / G
constexpr int KG   = 56;   // K * G
constexpr int LT   = 64;   // L-tile per workgroup

// LDS layout (floats). xs rows are x channels; interior cols at offset 4
// (16B aligned), halo cols j=-3..-1 at 1..3, right halo j=64..66 at 68..70.
constexpr int XS_STRIDE = 76;                 // multiple of 4; 76 mod 64 = 12
constexpr int XS_FLOATS = Cn * XS_STRIDE;     // 38912
constexpr int RS_STRIDE = 66;                 // de-aliases the M / M+8 C/D rows
constexpr int RS_OFF    = XS_FLOATS;
constexpr int RS_FLOATS = Cr * RS_STRIDE;     // 8448
constexpr int SP_OFF    = RS_OFF + RS_FLOATS;
constexpr int SP_FLOATS = KG * RS_STRIDE;     // 3696
constexpr int SH_FLOATS = SP_OFF + SP_FLOATS; // 51056
constexpr size_t SH_BYTES = size_t(SH_FLOATS) * sizeof(float); // 204224 B
} // namespace

__global__ __launch_bounds__(256) void involution1d_fused(
    const float* __restrict__ x,   // (B, C, L)
    const float* __restrict__ W0,  // (Cr, C)
    const float* __restrict__ b0,  // (Cr)
    const float* __restrict__ W1,  // (KG, Cr)
    const float* __restrict__ b1,  // (KG)
    float* __restrict__ out)       // (B, C, L)
{
    extern __shared__ float sh[];
    float* rs = sh + RS_OFF;   // reduced (Cr x 64), stride 66
    float* sp = sh + SP_OFF;   // spanned (KG x 64), stride 66

    const int tid  = threadIdx.x;
    const int l0   = blockIdx.x * LT;
    const int bb   = blockIdx.y;
    const int w    = tid >> 5;        // wave 0..7
    const int lane = tid & 31;
    const int m    = lane & 15;       // row/col within 16
    const int hi   = lane >> 4;       // half-wave select

    const float* xb = x + (size_t)bb * Cn * Ln;

    // ---------------- Stage A: stage x tile (with halo) into LDS -----------
    // Interior: 512 rows x 16 float4 (aligned, coalesced).
    for (int i = tid; i < Cn * (LT / 4); i += 256) {
        const int c = i >> 4;
        const int q = (i & 15) << 2;
        const float4 v = *(const float4*)(xb + c * Ln + l0 + q);
        *(float4*)&sh[c * XS_STRIDE + 4 + q] = v;
    }
    // Halo: 6 columns per row, zero-padded at the L edges.
    for (int i = tid; i < Cn * 6; i += 256) {
        const int c = i / 6;
        const int t = i - c * 6;
        const int j = (t < 3) ? (t - 3) : (t + 61);     // -3..-1, 64..66
        const int gl = l0 + j;
        sh[c * XS_STRIDE + 4 + j] =
            (gl >= 0 && gl < Ln) ? xb[c * Ln + gl] : 0.0f;
    }
    __syncthreads();

    // ---------------- Stage B: reduced = relu(W0 @ x + b0) via WMMA --------
    // Wave w owns reduced rows 16w..16w+15, all four 16-col tiles.
    {
        v8f a0 = {}, a1 = {}, a2 = {}, a3 = {};
        const float* w0row = W0 + (16 * w + m) * Cn;  // A: 16x4 fp32 layout
        #pragma unroll 4
        for (int kk = 0; kk < Cn; kk += 4) {
            const int kb = kk + 2 * hi;               // lanes 0-15: K=kk,kk+1
            v2f af;                                   // lanes16-31: K=kk+2,kk+3
            af.x = w0row[kb];
            af.y = w0row[kb + 1];
            const float* r0 = &sh[kb * XS_STRIDE + 4 + m];
            const float* r1 = r0 + XS_STRIDE;
            v2f bf0, bf1, bf2, bf3;
            bf0.x = r0[0];  bf0.y = r1[0];
            bf1.x = r0[16]; bf1.y = r1[16];
            bf2.x = r0[32]; bf2.y = r1[32];
            bf3.x = r0[48]; bf3.y = r1[48];
            a0 = wmma4(af, bf0, a0);
            a1 = wmma4(af, bf1, a1);
            a2 = wmma4(af, bf2, a2);
            a3 = wmma4(af, bf3, a3);
        }
        // Epilogue: C/D layout -> row = 16w + v + 8*hi, col = 16*ct + m.
        #pragma unroll
        for (int v = 0; v < 8; ++v) {
            const int row  = 16 * w + v + 8 * hi;
            const float bs = b0[row];
            float* dst = &rs[row * RS_STRIDE + m];
            dst[0]  = fmaxf(a0[v] + bs, 0.0f);
            dst[16] = fmaxf(a1[v] + bs, 0.0f);
            dst[32] = fmaxf(a2[v] + bs, 0.0f);
            dst[48] = fmaxf(a3[v] + bs, 0.0f);
        }
    }
    __syncthreads();

    // ---------------- Stage C: spanned = W1 @ reduced + b1 via WMMA --------
    // 64-padded rows: 4 row-tiles x 4 col-tiles; wave w -> row-tile w>>1,
    // col-tiles 2*(w&1), 2*(w&1)+1 (shared A-fragment).
    {
        const int rt  = w >> 1;
        const int ct0 = (w & 1) * 2;
        const int arow = 16 * rt + m;
        const bool av  = (arow < KG);
        const float* w1row = W1 + (av ? arow : 0) * Cr;
        v8f a0 = {}, a1 = {};
        #pragma unroll 4
        for (int kk = 0; kk < Cr; kk += 4) {
            const int kb = kk + 2 * hi;
            v2f af;
            af.x = av ? w1row[kb]     : 0.0f;
            af.y = av ? w1row[kb + 1] : 0.0f;
            const float* r0 = &rs[kb * RS_STRIDE + 16 * ct0 + m];
            const float* r1 = r0 + RS_STRIDE;
            v2f bf0, bf1;
            bf0.x = r0[0];  bf0.y = r1[0];
            bf1.x = r0[16]; bf1.y = r1[16];
            a0 = wmma4(af, bf0, a0);
            a1 = wmma4(af, bf1, a1);
        }
        #pragma unroll
        for (int v = 0; v < 8; ++v) {
            const int row = 16 * rt + v + 8 * hi;
            if (row < KG) {
                const float bs = b1[row];
                float* dst = &sp[row * RS_STRIDE + 16 * ct0 + m];
                dst[0]  = a0[v] + bs;
                dst[16] = a1[v] + bs;
            }
        }
    }
    __syncthreads();

    // ---------------- Stage D: per-pixel 7-tap involution ------------------
    // out[b, g*64+cg, l0+l] = sum_k sp[g*7+k][l] * x[b, ch, l0+l+k-3]
    {
        const int l  = tid & 63;
        const int cb = tid >> 6;                       // 0..3 channel phase
        float* outp = out + (size_t)bb * Cn * Ln + l0 + l;
        for (int g = 0; g < Gn; ++g) {
            float kern[Kk];
            #pragma unroll
            for (int q = 0; q < Kk; ++q)
                kern[q] = sp[(g * Kk + q) * RS_STRIDE + l];
            #pragma unroll 4
            for (int ii = 0; ii < 16; ++ii) {
                const int ch = g * Cg + cb + 4 * ii;
                const float* xr = &sh[ch * XS_STRIDE + 4 + l];
                float s = 0.0f;
                #pragma unroll
                for (int q = 0; q < Kk; ++q)
                    s = fmaf(kern[q], xr[q - 3], s);
                outp[(size_t)ch * Ln] = s;
            }
        }
    }
}

extern "C" void kernel_launch(void* const* d_in, const int* in_sizes, int n_in,
                              void* d_out, int out_size, void* d_ws, size_t ws_size,
                              hipStream_t stream) {
    (void)in_sizes; (void)n_in; (void)d_ws; (void)ws_size; (void)out_size;
    const float* x  = (const float*)d_in[0];
    const float* W0 = (const float*)d_in[1];
    const float* b0 = (const float*)d_in[2];
    const float* W1 = (const float*)d_in[3];
    const float* b1 = (const float*)d_in[4];
    float* out = (float*)d_out;

    dim3 grid(Ln / LT, Bn, 1);   // 64 x 8 workgroups
    dim3 block(256, 1, 1);       // 8 waves (wave32)
    involution1d_fused<<<grid, block, SH_BYTES, stream>>>(x, W0, b0, W1, b1, out);
}